// AmortizedCPTensor_23115513987494
// MI455X (gfx1250) — compile-verified
//
#include <hip/hip_runtime.h>
#include <hip/hip_bf16.h>
#include <math.h>

// Shapes (fixed by the reference): B=128, I=2000, K=512, R=64, H=128.
#define BB 128
#define II 2000
#define KK 512
#define RR 64
#define HH 128

typedef __attribute__((ext_vector_type(2))) float v2f;
typedef __attribute__((ext_vector_type(8))) float v8f;

// ---------------------------------------------------------------------------
// Kernel 1: encoder + reparameterization (tiny: ~65 MFLOP).
// One block per batch row b; 128 threads compute h[b, 0..127]; threads 0..63
// then compute mu/logvar/z for r = 0..63.
// ---------------------------------------------------------------------------
__global__ __launch_bounds__(128) void encoder_kernel(
    const float* __restrict__ x, const float* __restrict__ W1,
    const float* __restrict__ b1, const float* __restrict__ W_mu,
    const float* __restrict__ b_mu, const float* __restrict__ W_lv,
    const float* __restrict__ b_lv, const float* __restrict__ eps,
    float* __restrict__ mu_out, float* __restrict__ lv_out,
    float* __restrict__ z_out)
{
    const int b = blockIdx.x;
    const int t = threadIdx.x;  // 0..127

    __shared__ float sh_h[HH];

    // h[b,t] = relu(sum_i x[b,i] * W1[i,t] + b1[t]); W1 access coalesced in t.
    float acc = b1[t];
    const float* xr = x + (size_t)b * II;
    for (int i = 0; i < II; ++i) {
        acc = fmaf(xr[i], W1[(size_t)i * HH + t], acc);
    }
    sh_h[t] = fmaxf(acc, 0.0f);
    __syncthreads();

    if (t < RR) {
        float m  = b_mu[t];
        float lv = b_lv[t];
        for (int hh = 0; hh < HH; ++hh) {
            const float hv = sh_h[hh];
            m  = fmaf(hv, W_mu[(size_t)hh * RR + t], m);
            lv = fmaf(hv, W_lv[(size_t)hh * RR + t], lv);
        }
        const float sd = expf(0.5f * lv);
        const float zz = fmaf(eps[(size_t)b * RR + t], sd, m);
        mu_out[(size_t)b * RR + t] = m;
        lv_out[(size_t)b * RR + t] = lv;
        z_out [(size_t)b * RR + t] = zz;
    }
}

// ---------------------------------------------------------------------------
// Kernel 2: CP decode  X_hat[b,i,k] = sum_r z[b,r]*A[i,r]*C[k,r]
// Cast as flat GEMM:  ZA[b*I+i, r] = z[b,r]*A[i,r];  X_hat = ZA @ C^T.
// f32 WMMA 16x16x4, K-dim = 64 -> 16 WMMA per accumulator.
// Block: 256 threads (8 waves). Wave w handles M-tile (blockIdx.x*8 + w) of
// batch blockIdx.y (I = 2000 = 125 tiles of 16 rows), iterating the 32
// N-tiles in pairs (two accumulators for ILP).
//
// Fragment layouts per CDNA5 ISA 7.12.2 (wave32):
//   A 16x4 f32:  lane = half*16 + m (m=0..15); VGPR0 = K=half*2, VGPR1 = K=half*2+1
//   B  4x16 f32: lane = half*16 + n (n=0..15); VGPR0 = K=half*2, VGPR1 = K=half*2+1
//   C/D 16x16:   VGPR j: lanes 0-15 -> (row j,   col lane)
//                        lanes 16-31 -> (row j+8, col lane-16)
// ---------------------------------------------------------------------------
__global__ __launch_bounds__(256) void decode_kernel(
    const float* __restrict__ z,    // [B, R]
    const float* __restrict__ A,    // [I, R]
    const float* __restrict__ C,    // [K, R]
    float* __restrict__ out)        // [B, I, K]
{
    const int b    = blockIdx.y;
    const int wave = threadIdx.x >> 5;
    const int lane = threadIdx.x & 31;

    const int mtile = blockIdx.x * 8 + wave;
    if (mtile >= II / 16) return;  // 125 M-tiles per batch; no barriers below

    const int i_base = mtile * 16;
    const int m      = lane & 15;  // row (A-frag) / col (B-frag, C-frag) index
    const int half   = lane >> 4;  // selects K pair {0,1} vs {2,3} within 16x16x4

    // Build the full K=64 slab of A-fragments once: za[c] covers r = 4c..4c+3.
    const float* Arow = A + (size_t)(i_base + m) * RR;
    const float* zrow = z + (size_t)b * RR;
    v2f za[16];
#pragma unroll
    for (int c = 0; c < 16; ++c) {
        const int r = c * 4 + half * 2;
        const v2f a  = *(const v2f*)(Arow + r);
        const v2f zv = *(const v2f*)(zrow + r);
        v2f p; p.x = a.x * zv.x; p.y = a.y * zv.y;
        za[c] = p;
    }

    float* outb = out + ((size_t)b * II + i_base) * KK;

#pragma unroll 1
    for (int nt2 = 0; nt2 < 16; ++nt2) {
        const int k0a = nt2 * 32;
        const int k0b = k0a + 16;
        const float* Ca = C + (size_t)(k0a + m) * RR;  // B-frag: col n = lane%16
        const float* Cb = C + (size_t)(k0b + m) * RR;

        v8f acc0 = {};
        v8f acc1 = {};
#pragma unroll
        for (int c = 0; c < 16; ++c) {
            const int r = c * 4 + half * 2;
            const v2f bf0 = *(const v2f*)(Ca + r);  // C^T[r..r+1, k0a+n]
            const v2f bf1 = *(const v2f*)(Cb + r);
            acc0 = __builtin_amdgcn_wmma_f32_16x16x4_f32(
                false, za[c], false, bf0, (short)0, acc0, false, false);
            acc1 = __builtin_amdgcn_wmma_f32_16x16x4_f32(
                false, za[c], false, bf1, (short)0, acc1, false, false);
        }

        // Store D: VGPR j -> row (j + half*8), col (k0 + m). Lanes 0-15 write
        // 16 consecutive floats (64B), likewise lanes 16-31.
#pragma unroll
        for (int j = 0; j < 8; ++j) {
            const size_t row = (size_t)(j + half * 8);
            outb[row * KK + k0a + m] = acc0[j];
            outb[row * KK + k0b + m] = acc1[j];
        }
    }
}

// ---------------------------------------------------------------------------
// Launch
// ---------------------------------------------------------------------------
extern "C" void kernel_launch(void* const* d_in, const int* in_sizes, int n_in,
                              void* d_out, int out_size, void* d_ws, size_t ws_size,
                              hipStream_t stream) {
    const float* x    = (const float*)d_in[0];
    const float* W1   = (const float*)d_in[1];
    const float* b1   = (const float*)d_in[2];
    const float* W_mu = (const float*)d_in[3];
    const float* b_mu = (const float*)d_in[4];
    const float* W_lv = (const float*)d_in[5];
    const float* b_lv = (const float*)d_in[6];
    const float* A    = (const float*)d_in[7];
    const float* Cm   = (const float*)d_in[8];
    const float* eps  = (const float*)d_in[9];

    float* out = (float*)d_out;
    const size_t BIK = (size_t)BB * II * KK;   // 131,072,000
    float* mu_out = out + BIK;                 // [B, R]
    float* lv_out = mu_out + (size_t)BB * RR;  // [B, R]
    float* zbuf   = (float*)d_ws;              // [B, R] scratch (32 KB)

    encoder_kernel<<<dim3(BB), dim3(HH), 0, stream>>>(
        x, W1, b1, W_mu, b_mu, W_lv, b_lv, eps, mu_out, lv_out, zbuf);

    // 125 M-tiles per batch -> 16 blocks of 8 waves (last block 5 active waves)
    decode_kernel<<<dim3(16, BB), dim3(256), 0, stream>>>(zbuf, A, Cm, out);
}